// FeatureAttention_48077863911955
// MI455X (gfx1250) — compile-verified
//
#include <hip/hip_runtime.h>
#include <hip/hip_bf16.h>

typedef __attribute__((ext_vector_type(2))) float f32x2;
typedef __attribute__((ext_vector_type(8))) float f32x8;

constexpr int N_NODES  = 500000;
constexpr int HIDDEN   = 128;
constexpr int REDUCED  = 64;
constexpr int N_GRAPHS = 4096;
constexpr int NPB      = 1024;   // nodes per pooling block

// ---------------------------------------------------------------- zero scratch
__global__ __launch_bounds__(256) void fa_zero_kernel(float* __restrict__ p, int n) {
    int i = blockIdx.x * 256 + threadIdx.x;
    if (i < n) p[i] = 0.0f;
}

// ---------------------------------------------------------------- segment-sum pooling
// batch is sorted -> contiguous runs. Each block owns NPB contiguous nodes,
// stages batch ids in LDS, threads keep a float4 run-accumulator and flush via
// atomicAdd only at run boundaries. Each wave reads one full 512B node row per
// iteration (32 lanes x float4) -> coalesced HBM streaming.
__global__ __launch_bounds__(256) void fa_pool_kernel(const float4* __restrict__ x,
                                                      const int*    __restrict__ batch,
                                                      float*        __restrict__ pooled) {
    __shared__ int sbatch[NPB];
    const int start = blockIdx.x * NPB;
    const int count = min(NPB, N_NODES - start);
    for (int i = threadIdx.x; i < count; i += 256) sbatch[i] = batch[start + i];
    __syncthreads();

    const int lane = threadIdx.x & 31;   // float4 channel index (0..31)
    const int sub  = threadIdx.x >> 5;   // which of 8 interleaved node streams

    float4 acc = make_float4(0.f, 0.f, 0.f, 0.f);
    int cur = -1;
    for (int j = sub; j < count; j += 8) {
        const int b = sbatch[j];
        if (b != cur) {
            if (cur >= 0) {
                float* dst = pooled + (size_t)cur * HIDDEN + lane * 4;
                atomicAdd(dst + 0, acc.x);
                atomicAdd(dst + 1, acc.y);
                atomicAdd(dst + 2, acc.z);
                atomicAdd(dst + 3, acc.w);
            }
            cur = b;
            acc = make_float4(0.f, 0.f, 0.f, 0.f);
        }
        const float4 v = x[(size_t)(start + j) * 32 + lane];
        acc.x += v.x; acc.y += v.y; acc.z += v.z; acc.w += v.w;
    }
    if (cur >= 0) {
        float* dst = pooled + (size_t)cur * HIDDEN + lane * 4;
        atomicAdd(dst + 0, acc.x);
        atomicAdd(dst + 1, acc.y);
        atomicAdd(dst + 2, acc.z);
        atomicAdd(dst + 3, acc.w);
    }
}

// ---------------------------------------------------------------- SE-MLP via fp32 WMMA
// One wave computes a 16-row tile: relu(pooled@W1+b1) then sigmoid(h@W2+b2).
// Uses V_WMMA_F32_16X16X4_F32 (fp32 in/out, no precision loss).
__global__ __launch_bounds__(32) void fa_mlp_kernel(const float* __restrict__ pooled,
                                                    const float* __restrict__ W1,
                                                    const float* __restrict__ b1,
                                                    const float* __restrict__ W2,
                                                    const float* __restrict__ b2,
                                                    float*       __restrict__ y) {
    __shared__ float hlds[16 * REDUCED];   // 4 KB: h tile, re-laid out D->A via LDS

    const int lane = threadIdx.x;          // 0..31
    const int m    = lane & 15;            // A row / B,C,D column within 16-tile
    const int kh   = lane >> 4;            // K-half selector (0: K=0,1  1: K=2,3)
    const int m0   = blockIdx.x * 16;      // global row base

    // -------- GEMM1: [16x128] x [128x64] + b1, ReLU --------
    f32x8 acc[4];
    for (int nt = 0; nt < 4; ++nt) {
        const float bv = b1[nt * 16 + m];
        for (int r = 0; r < 8; ++r) acc[nt][r] = bv;
    }
    for (int k0 = 0; k0 < HIDDEN; k0 += 4) {
        f32x2 a;
        const float* ap = pooled + (size_t)(m0 + m) * HIDDEN + k0 + 2 * kh;
        a.x = ap[0];
        a.y = ap[1];
#pragma unroll
        for (int nt = 0; nt < 4; ++nt) {
            f32x2 b;
            b.x = W1[(size_t)(k0 + 2 * kh)     * REDUCED + nt * 16 + m];
            b.y = W1[(size_t)(k0 + 2 * kh + 1) * REDUCED + nt * 16 + m];
            acc[nt] = __builtin_amdgcn_wmma_f32_16x16x4_f32(
                false, a, false, b, (short)0, acc[nt], false, false);
        }
    }
    // D layout: vgpr r, lanes 0-15 -> row r; lanes 16-31 -> row r+8; col = lane%16
    for (int nt = 0; nt < 4; ++nt)
        for (int r = 0; r < 8; ++r)
            hlds[(r + 8 * kh) * REDUCED + nt * 16 + m] = fmaxf(acc[nt][r], 0.0f);
    __syncthreads();

    // -------- GEMM2: [16x64] x [64x128] + b2, sigmoid --------
    for (int nt = 0; nt < 8; ++nt) {
        f32x8 c;
        const float bv = b2[nt * 16 + m];
        for (int r = 0; r < 8; ++r) c[r] = bv;
#pragma unroll
        for (int k0 = 0; k0 < REDUCED; k0 += 4) {
            f32x2 a, b;
            a.x = hlds[m * REDUCED + k0 + 2 * kh];
            a.y = hlds[m * REDUCED + k0 + 2 * kh + 1];
            b.x = W2[(size_t)(k0 + 2 * kh)     * HIDDEN + nt * 16 + m];
            b.y = W2[(size_t)(k0 + 2 * kh + 1) * HIDDEN + nt * 16 + m];
            c = __builtin_amdgcn_wmma_f32_16x16x4_f32(
                false, a, false, b, (short)0, c, false, false);
        }
        for (int r = 0; r < 8; ++r) {
            const float v = 1.0f / (1.0f + __expf(-c[r]));
            y[(size_t)(m0 + r + 8 * kh) * HIDDEN + nt * 16 + m] = v;
        }
    }
}

// ---------------------------------------------------------------- gather + modulate
// One wave per node: batch[node] is wave-uniform (scalar load); float4 streaming.
__global__ __launch_bounds__(256) void fa_mod_kernel(const float4* __restrict__ x,
                                                     const int*    __restrict__ batch,
                                                     const float4* __restrict__ y,
                                                     float4*       __restrict__ out) {
    const size_t idx = (size_t)blockIdx.x * 256 + threadIdx.x;
    if (idx >= (size_t)N_NODES * 32) return;
    const size_t node = idx >> 5;
    const int    c4   = (int)(idx & 31);
    const int    b    = batch[node];
    const float4 xv   = x[idx];
    const float4 yv   = y[(size_t)b * 32 + c4];
    float4 o;
    o.x = xv.x * yv.x;
    o.y = xv.y * yv.y;
    o.z = xv.z * yv.z;
    o.w = xv.w * yv.w;
    out[idx] = o;
}

extern "C" void kernel_launch(void* const* d_in, const int* in_sizes, int n_in,
                              void* d_out, int out_size, void* d_ws, size_t ws_size,
                              hipStream_t stream) {
    const float* x     = (const float*)d_in[0];
    const int*   batch = (const int*)  d_in[1];
    // d_in[2] = size scalar (compile-time N_GRAPHS here)
    const float* W1    = (const float*)d_in[3];
    const float* b1    = (const float*)d_in[4];
    const float* W2    = (const float*)d_in[5];
    const float* b2    = (const float*)d_in[6];
    float*       out   = (float*)d_out;

    float* pooled = (float*)d_ws;                              // 2 MB
    float* y      = (float*)d_ws + (size_t)N_GRAPHS * HIDDEN;  // 2 MB

    const int pooledN = N_GRAPHS * HIDDEN;
    fa_zero_kernel<<<(pooledN + 255) / 256, 256, 0, stream>>>(pooled, pooledN);

    fa_pool_kernel<<<(N_NODES + NPB - 1) / NPB, 256, 0, stream>>>(
        (const float4*)x, batch, pooled);

    fa_mlp_kernel<<<N_GRAPHS / 16, 32, 0, stream>>>(pooled, W1, b1, W2, b2, y);

    const long long total = (long long)N_NODES * 32;
    fa_mod_kernel<<<(unsigned)((total + 255) / 256), 256, 0, stream>>>(
        (const float4*)x, batch, (const float4*)y, (float4*)out);
}